// Mask2ControlPoints_66236985639391
// MI455X (gfx1250) — compile-verified
//
#include <hip/hip_runtime.h>
#include <math.h>

typedef __attribute__((ext_vector_type(2))) float v2f;
typedef __attribute__((ext_vector_type(8))) float v8f;

namespace {
constexpr int Bc = 64;     // batch
constexpr int Sc = 50;     // seq len
constexpr int HIDc = 512;
constexpr int Vc = 4096;   // vocab (start token id == Vc)
constexpr int G4 = 2048;   // 4*HID
constexpr int Hc = 256, Wc = 256, PH = 128, PW = 128;
}

// ---------------------------------------------------------------------------
// conv1 (1->16, 3x3 SAME) + ReLU + 2x2 maxpool, fused. One thread per pooled
// output position, all 16 output channels. max(relu(s+b)) == relu(max(s)+b).
// ---------------------------------------------------------------------------
__global__ __launch_bounds__(256) void conv1_pool_kernel(
    const float* __restrict__ mask, const float* __restrict__ w,
    const float* __restrict__ bias, float* __restrict__ out)
{
  __shared__ float sw[16 * 9];
  __shared__ float sb[16];
  const int tid = threadIdx.x;
  if (tid < 144) sw[tid] = w[tid];
  if (tid < 16)  sb[tid] = bias[tid];
  __syncthreads();

  const int gid = blockIdx.x * 256 + tid;        // 64*128*128 total
  const int b   = gid >> 14;
  const int pos = gid & 16383;
  const int ph  = pos >> 7, pw = pos & 127;

  const float* img = mask + (size_t)b * (Hc * Wc);
  float patch[4][4];
#pragma unroll
  for (int dy = 0; dy < 4; ++dy) {
    const int iy = 2 * ph + dy - 1;
#pragma unroll
    for (int dx = 0; dx < 4; ++dx) {
      const int ix = 2 * pw + dx - 1;
      const bool ok = (iy >= 0) && (iy < Hc) && (ix >= 0) && (ix < Wc);
      patch[dy][dx] = ok ? img[iy * Wc + ix] : 0.0f;
    }
  }

  float* ob = out + (size_t)b * 16 * (PH * PW) + ph * PW + pw;
#pragma unroll
  for (int oc = 0; oc < 16; ++oc) {
    const float* wk = &sw[oc * 9];
    float mx = -INFINITY;
#pragma unroll
    for (int py = 0; py < 2; ++py)
#pragma unroll
      for (int px = 0; px < 2; ++px) {
        float s = 0.0f;
#pragma unroll
        for (int ky = 0; ky < 3; ++ky)
#pragma unroll
          for (int kx = 0; kx < 3; ++kx)
            s = fmaf(wk[ky * 3 + kx], patch[py + ky][px + kx], s);
        mx = fmaxf(mx, s);
      }
    ob[oc * (PH * PW)] = fmaxf(mx + sb[oc], 0.0f);
  }
}

// ---------------------------------------------------------------------------
// conv2 (16->32, 3x3 SAME) + ReLU + global average pool.
// One block per (b, oc); block reduction over 128x128 positions.
// ---------------------------------------------------------------------------
__global__ __launch_bounds__(256) void conv2_gap_kernel(
    const float* __restrict__ pool1, const float* __restrict__ w,
    const float* __restrict__ bias, float* __restrict__ xout)
{
  const int b  = blockIdx.x >> 5;
  const int oc = blockIdx.x & 31;
  __shared__ float sw[16 * 9];
  __shared__ float red[256];
  const int tid = threadIdx.x;
  if (tid < 144) sw[tid] = w[oc * 144 + tid];
  __syncthreads();

  const float bia = bias[oc];
  const float* src = pool1 + (size_t)b * 16 * (PH * PW);
  float acc = 0.0f;
  for (int pos = tid; pos < PH * PW; pos += 256) {
    const int y = pos >> 7, x = pos & 127;
    float s = bia;
    for (int ic = 0; ic < 16; ++ic) {
      const float* ch = src + ic * (PH * PW);
      const float* wk = &sw[ic * 9];
#pragma unroll
      for (int ky = 0; ky < 3; ++ky) {
        const int iy = y + ky - 1;
        if (iy < 0 || iy >= PH) continue;
#pragma unroll
        for (int kx = 0; kx < 3; ++kx) {
          const int ix = x + kx - 1;
          if (ix < 0 || ix >= PW) continue;
          s = fmaf(wk[ky * 3 + kx], ch[iy * PW + ix], s);
        }
      }
    }
    acc += fmaxf(s, 0.0f);
  }
  red[tid] = acc;
  __syncthreads();
  for (int st = 128; st > 0; st >>= 1) {
    if (tid < st) red[tid] += red[tid + st];
    __syncthreads();
  }
  if (tid == 0) xout[b * 32 + oc] = red[0] * (1.0f / (PH * PW));
}

// ---------------------------------------------------------------------------
// h0 = x @ feat_w.T + feat_b ; also seed hbuf=h0, cbuf=0
// ---------------------------------------------------------------------------
__global__ __launch_bounds__(256) void feat_kernel(
    const float* __restrict__ x32, const float* __restrict__ fw,
    const float* __restrict__ fb, float* __restrict__ h0,
    float* __restrict__ hbuf, float* __restrict__ cbuf)
{
  const int idx = blockIdx.x * 256 + threadIdx.x;   // 64*512
  const int b = idx >> 9, j = idx & 511;
  float s = fb[j];
  const float* xr = x32 + b * 32;
  const float* wr = fw + j * 32;
#pragma unroll
  for (int k = 0; k < 32; ++k) s = fmaf(xr[k], wr[k], s);
  h0[idx] = s; hbuf[idx] = s; cbuf[idx] = 0.0f;
}

// ctx = h0 @ ctx_w.T + ctx_b
__global__ __launch_bounds__(256) void ctx_kernel(
    const float* __restrict__ h0, const float* __restrict__ cw,
    const float* __restrict__ cb, float* __restrict__ ctx)
{
  const int idx = blockIdx.x * 256 + threadIdx.x;
  const int b = idx >> 9, j = idx & 511;
  float s = cb[j];
  const float* hr = h0 + b * HIDc;
  const float* wr = cw + (size_t)j * HIDc;
  for (int k = 0; k < HIDc; ++k) s = fmaf(hr[k], wr[k], s);
  ctx[idx] = s;
}

__global__ void bias_comb_kernel(const float* __restrict__ a,
                                 const float* __restrict__ b,
                                 float* __restrict__ o)
{
  const int i = blockIdx.x * 256 + threadIdx.x;   // 2048
  o[i] = a[i] + b[i];
}

// ---------------------------------------------------------------------------
// fp32 WMMA GEMM:  Out = A(MxK) @ W(NxK)^T  (+ bias / + addend), tiled 64x64
// per block, 8 waves, each wave -> 16x32 via two V_WMMA_F32_16X16X4_F32 accs.
// MODE 0 (XGATES): A row m gathered as tok_emb[tok(s,b)] + ctx[b]; +bias
// MODE 1 (HGEMM):  plain A; + addend matrix (precomputed x-gates for step)
// MODE 2 (HEAD):   plain A; +bias; output remapped (s,b,n) -> (b,s,n)
// All dims divide tile sizes exactly -> EXEC all-1s at every WMMA.
// ---------------------------------------------------------------------------
enum GemmMode { XGATES = 0, HGEMM = 1, HEAD = 2 };

template <int MODE>
__global__ __launch_bounds__(256) void gemm_wmma_kernel(
    const float* __restrict__ A, const float* __restrict__ Wt,
    const float* __restrict__ bias, const float* __restrict__ addend,
    const float* __restrict__ tok_emb, const int* __restrict__ tgt,
    const float* __restrict__ ctx, float* __restrict__ Out,
    int M, int N, int K)
{
  __shared__ float As[64 * 17];   // pad stride 17: coprime with 64 banks
  __shared__ float Ws[64 * 17];

  const int tid  = threadIdx.x;
  const int lane = tid & 31;
  const int wave = tid >> 5;          // 0..7
  const int half = lane >> 4;         // lane-half -> K pair select
  const int l16  = lane & 15;
  const int mOff = (wave >> 1) * 16;  // 0,16,32,48
  const int nOff = (wave & 1) * 32;   // 0,32
  const int m0 = blockIdx.y * 64;
  const int n0 = blockIdx.x * 64;

  // cooperative staging role: each thread owns 4 consecutive K of one row
  const int r = tid >> 2;             // 0..63 (row within tile)
  const int q = (tid & 3) * 4;        // 0,4,8,12 (K offset)

  const float* aRow;
  const float* ctxRow = nullptr;
  if (MODE == XGATES) {
    const int m = m0 + r;
    const int s = m >> 6, b = m & 63;
    const int tok = (s == 0) ? Vc : tgt[b * Sc + (s - 1)];
    aRow   = tok_emb + (size_t)tok * K;
    ctxRow = ctx + (size_t)b * K;
  } else {
    aRow = A + (size_t)(m0 + r) * K;
  }
  const float* wRow = Wt + (size_t)(n0 + r) * K;

  v8f c0 = {}; v8f c1 = {};

  for (int k0 = 0; k0 < K; k0 += 16) {
#pragma unroll
    for (int j = 0; j < 4; ++j) {
      float av = aRow[k0 + q + j];
      if (MODE == XGATES) av += ctxRow[k0 + q + j];
      As[r * 17 + q + j] = av;
      Ws[r * 17 + q + j] = wRow[k0 + q + j];
    }
    if (k0 + 16 < K) {                 // global_prefetch_b8 of next K chunk
      __builtin_prefetch(&aRow[k0 + 16 + q], 0, 1);
      __builtin_prefetch(&wRow[k0 + 16 + q], 0, 1);
    }
    __syncthreads();

#pragma unroll
    for (int kk = 0; kk < 4; ++kk) {
      const int kb = kk * 4 + half * 2;
      v2f a, b0, b1;
      // A 16x4: lanes 0-15 M=0..15 K={0,1}; lanes 16-31 K={2,3}
      a.x  = As[(mOff + l16) * 17 + kb];
      a.y  = As[(mOff + l16) * 17 + kb + 1];
      // B 4x16 mirrored (B[k][n] = W[n][k])
      b0.x = Ws[(nOff + l16) * 17 + kb];
      b0.y = Ws[(nOff + l16) * 17 + kb + 1];
      b1.x = Ws[(nOff + 16 + l16) * 17 + kb];
      b1.y = Ws[(nOff + 16 + l16) * 17 + kb + 1];
      c0 = __builtin_amdgcn_wmma_f32_16x16x4_f32(false, a, false, b0,
                                                 (short)0, c0, false, false);
      c1 = __builtin_amdgcn_wmma_f32_16x16x4_f32(false, a, false, b1,
                                                 (short)0, c1, false, false);
    }
    __syncthreads();
  }

  // C/D layout: VGPR r -> M = r + 8*laneHalf, N = lane&15
#pragma unroll
  for (int rr = 0; rr < 8; ++rr) {
    const int m = m0 + mOff + half * 8 + rr;
#pragma unroll
    for (int t = 0; t < 2; ++t) {
      const int n = n0 + nOff + t * 16 + l16;
      float v = (t == 0) ? c0[rr] : c1[rr];
      if (MODE == HGEMM) {
        v += addend[(size_t)m * N + n];
        Out[(size_t)m * N + n] = v;
      } else if (MODE == XGATES) {
        v += bias[n];
        Out[(size_t)m * N + n] = v;
      } else {  // HEAD: m = s*64 + b  ->  out[b][s][n]
        v += bias[n];
        const int s = m >> 6, b = m & 63;
        Out[((size_t)b * Sc + s) * N + n] = v;
      }
    }
  }
}

// ---------------------------------------------------------------------------
// Per-step LSTM elementwise: gates (64,2048) row-major, split i|f|g|o.
// ---------------------------------------------------------------------------
__global__ __launch_bounds__(256) void lstm_elem_kernel(
    const float* __restrict__ gates, float* __restrict__ hbuf,
    float* __restrict__ cbuf, float* __restrict__ h_all, int s)
{
  const int idx = blockIdx.x * 256 + threadIdx.x;   // 64*512
  const int b = idx >> 9, j = idx & 511;
  const float* g = gates + b * G4;
  const float gi = g[j], gf = g[512 + j], gg = g[1024 + j], go = g[1536 + j];
  const float si = 1.0f / (1.0f + __expf(-gi));
  const float sf = 1.0f / (1.0f + __expf(-gf));
  const float so = 1.0f / (1.0f + __expf(-go));
  const float c = sf * cbuf[idx] + si * tanhf(gg);
  const float h = so * tanhf(c);
  cbuf[idx] = c;
  hbuf[idx] = h;
  h_all[((size_t)s * Bc + b) * HIDc + j] = h;
}

// ---------------------------------------------------------------------------
extern "C" void kernel_launch(void* const* d_in, const int* in_sizes, int n_in,
                              void* d_out, int out_size, void* d_ws, size_t ws_size,
                              hipStream_t stream)
{
  const float* mask    = (const float*)d_in[0];
  const int*   target  = (const int*)  d_in[1];
  const float* conv1_w = (const float*)d_in[2];
  const float* conv1_b = (const float*)d_in[3];
  const float* conv2_w = (const float*)d_in[4];
  const float* conv2_b = (const float*)d_in[5];
  const float* feat_w  = (const float*)d_in[6];
  const float* feat_b  = (const float*)d_in[7];
  const float* ctx_w   = (const float*)d_in[8];
  const float* ctx_b   = (const float*)d_in[9];
  const float* tok_emb = (const float*)d_in[10];
  const float* w_ih    = (const float*)d_in[11];
  const float* w_hh    = (const float*)d_in[12];
  const float* b_ih    = (const float*)d_in[13];
  const float* b_hh    = (const float*)d_in[14];
  const float* head_w  = (const float*)d_in[15];
  const float* head_b  = (const float*)d_in[16];
  float* out = (float*)d_out;

  // workspace layout (floats), ~101 MB total
  float* ws    = (float*)d_ws;
  float* pool1 = ws;                                   // 64*16*128*128
  float* xg    = pool1 + (size_t)Bc * 16 * PH * PW;    // 3200*2048
  float* h_all = xg    + (size_t)Sc * Bc * G4;         // 3200*512
  float* x32   = h_all + (size_t)Sc * Bc * HIDc;       // 64*32
  float* h0    = x32   + Bc * 32;                      // 64*512
  float* ctx   = h0    + Bc * HIDc;                    // 64*512
  float* hbuf  = ctx   + Bc * HIDc;                    // 64*512
  float* cbuf  = hbuf  + Bc * HIDc;                    // 64*512
  float* gates = cbuf  + Bc * HIDc;                    // 64*2048
  float* biasc = gates + Bc * G4;                      // 2048

  // --- encoder ---
  conv1_pool_kernel<<<(Bc * PH * PW) / 256, 256, 0, stream>>>(
      mask, conv1_w, conv1_b, pool1);
  conv2_gap_kernel<<<Bc * 32, 256, 0, stream>>>(pool1, conv2_w, conv2_b, x32);
  feat_kernel<<<(Bc * HIDc) / 256, 256, 0, stream>>>(x32, feat_w, feat_b,
                                                     h0, hbuf, cbuf);
  ctx_kernel<<<(Bc * HIDc) / 256, 256, 0, stream>>>(h0, ctx_w, ctx_b, ctx);
  bias_comb_kernel<<<G4 / 256, 256, 0, stream>>>(b_ih, b_hh, biasc);

  // --- hoisted input-side GEMM for all 50 steps: (3200,512)x(512,2048) ---
  gemm_wmma_kernel<XGATES><<<dim3(G4 / 64, (Sc * Bc) / 64), 256, 0, stream>>>(
      nullptr, w_ih, biasc, nullptr, tok_emb, target, ctx, xg,
      Sc * Bc, G4, HIDc);

  // --- serial recurrence: only h @ W_hh.T per step ---
  for (int s = 0; s < Sc; ++s) {
    gemm_wmma_kernel<HGEMM><<<dim3(G4 / 64, 1), 256, 0, stream>>>(
        hbuf, w_hh, nullptr, xg + (size_t)s * Bc * G4,
        nullptr, nullptr, nullptr, gates, Bc, G4, HIDc);
    lstm_elem_kernel<<<(Bc * HIDc) / 256, 256, 0, stream>>>(
        gates, hbuf, cbuf, h_all, s);
  }

  // --- hoisted head GEMM over all steps: (3200,512)x(512,4096) ---
  gemm_wmma_kernel<HEAD><<<dim3(Vc / 64, (Sc * Bc) / 64), 256, 0, stream>>>(
      h_all, head_w, head_b, nullptr, nullptr, nullptr, nullptr, out,
      Sc * Bc, Vc, HIDc);
}